// QuantumForecaster_20272245637761
// MI455X (gfx1250) — compile-verified
//
#include <hip/hip_runtime.h>
#include <math.h>

typedef __attribute__((ext_vector_type(2))) float v2f;
typedef __attribute__((ext_vector_type(8))) float v8f;

#define NQ     8
#define QDIM   256        // 2^NQ
#define NL     4
#define WFEAT  50
#define BATCH  32768
#define TILE_M 64         // batch rows per workgroup in GEMM
#define KDIM   512        // realified K (re|im)
#define NDIM   512        // realified N (T_re|T_im)

// ---------------- Kernel 1: angles -> per-(batch,wire) cos/sin ----------------
// csbuf[b*16 + 2*w + 0] = cos(angle/2), +1 = sin(angle/2)
__global__ void __launch_bounds__(256)
qf_angles_kernel(const float* __restrict__ x,
                 const float* __restrict__ Wp,
                 const float* __restrict__ bp,
                 float* __restrict__ csbuf) {
  int gid = blockIdx.x * blockDim.x + threadIdx.x;   // BATCH*NQ threads
  int b = gid >> 3, w = gid & 7;
  if (b >= BATCH) return;
  const float* xr = x + b * WFEAT;
  const float* wr = Wp + w * WFEAT;
  float a = bp[w];
  #pragma unroll
  for (int i = 0; i < WFEAT; ++i) a = fmaf(xr[i], wr[i], a);
  float h = 0.5f * a;
  csbuf[b * 16 + w * 2 + 0] = __cosf(h);
  csbuf[b * 16 + w * 2 + 1] = __sinf(h);
}

// ---------------- Kernel 2: build realified circuit matrix B (512x512) --------
// One workgroup evolves basis column |j>; thread t holds amplitude of basis t.
// Wire w corresponds to bit (7-w) of the basis index (wire 0 = MSB).
__global__ void __launch_bounds__(QDIM)
qf_build_B_kernel(const float* __restrict__ qw,
                  float* __restrict__ Bmat) {
  __shared__ float bA_re[QDIM], bA_im[QDIM], bB_re[QDIM], bB_im[QDIM];
  int t   = threadIdx.x;
  int col = blockIdx.x;
  float *cur_re = bA_re, *cur_im = bA_im, *nxt_re = bB_re, *nxt_im = bB_im;
  cur_re[t] = (t == col) ? 1.0f : 0.0f;
  cur_im[t] = 0.0f;
  __syncthreads();

  for (int l = 0; l < NL; ++l) {
    // Rot(phi, theta, omega) = Rz(om) Ry(th) Rz(ph) on each wire
    for (int w = 0; w < NQ; ++w) {
      const float* q = qw + (l * NQ + w) * 3;
      float ph = q[0], th = q[1], om = q[2];
      float hs = 0.5f * (ph + om), hd = 0.5f * (ph - om);
      float ct = __cosf(0.5f * th), st = __sinf(0.5f * th);
      float chs = __cosf(hs), shs = __sinf(hs);
      float chd = __cosf(hd), shd = __sinf(hd);
      float u00r =  ct * chs, u00i = -ct * shs;
      float u01r = -st * chd, u01i = -st * shd;
      float u10r =  st * chd, u10i = -st * shd;
      float u11r =  ct * chs, u11i =  ct * shs;
      int mask = 1 << (7 - w);
      int pt = t ^ mask;
      float sr = cur_re[t],  si = cur_im[t];
      float pr = cur_re[pt], pi = cur_im[pt];
      float nr, ni;
      if (t & mask) {  // a0 = partner (bit=0), a1 = self : new = u10*a0 + u11*a1
        nr = u10r * pr - u10i * pi + u11r * sr - u11i * si;
        ni = u10r * pi + u10i * pr + u11r * si + u11i * sr;
      } else {         // new = u00*self + u01*partner
        nr = u00r * sr - u00i * si + u01r * pr - u01i * pi;
        ni = u00r * si + u00i * sr + u01r * pi + u01i * pr;
      }
      nxt_re[t] = nr; nxt_im[t] = ni;
      __syncthreads();
      float* tr = cur_re; cur_re = nxt_re; nxt_re = tr;
      float* ti = cur_im; cur_im = nxt_im; nxt_im = ti;
    }
    // CNOT ring with range r = l % 7 + 1
    int r = l % (NQ - 1) + 1;
    for (int w = 0; w < NQ; ++w) {
      int cm = 1 << (7 - w);
      int tm = 1 << (7 - ((w + r) % NQ));
      int src = (t & cm) ? (t ^ tm) : t;
      float nr = cur_re[src], ni = cur_im[src];
      nxt_re[t] = nr; nxt_im[t] = ni;
      __syncthreads();
      float* tr = cur_re; cur_re = nxt_re; nxt_re = tr;
      float* ti = cur_im; cur_im = nxt_im; nxt_im = ti;
    }
  }

  // U[t][col] = (re, im). Realified layout for T = A * Bmat, A = [S_re|S_im]:
  //   Bmat[j][n<256]        = Ur[n][j]      Bmat[j][n>=256]       = Ui[n-256][j]
  //   Bmat[j+256][n<256]    = -Ui[n][j]     Bmat[j+256][n>=256]   = Ur[n-256][j-]
  float re = cur_re[t], im = cur_im[t];
  Bmat[col * NDIM + t]                   =  re;
  Bmat[(col + QDIM) * NDIM + t]          = -im;
  Bmat[col * NDIM + QDIM + t]            =  im;
  Bmat[(col + QDIM) * NDIM + QDIM + t]   =  re;
}

// ---------------- Kernel 3: fp32 WMMA GEMM + sign-|.|^2 reduction -------------
__global__ void __launch_bounds__(256)
qf_gemm_kernel(const float* __restrict__ csbuf,
               const float* __restrict__ Bmat,
               float* __restrict__ out) {
  __shared__ float Asub[TILE_M * KDIM];   // 128 KB  (fits in 320 KB WGP LDS)
  __shared__ float csl[TILE_M * 16];      // 4 KB
  __shared__ float yacc[TILE_M];

  int tid  = threadIdx.x;
  int lane = tid & 31;
  int wave = tid >> 5;
  int m0   = blockIdx.x * TILE_M;

  for (int i = tid; i < TILE_M * 16; i += 256) csl[i] = csbuf[m0 * 16 + i];
  if (tid < TILE_M) yacc[tid] = 0.0f;
  __syncthreads();

  // Materialize A = [S_re | S_im] for 64 batch rows from product-state factors.
  // s[j] = (prod of cos/sin per wire-bit) * (-i)^popcount(j)
  for (int idx = tid; idx < TILE_M * KDIM; idx += 256) {
    int m = idx >> 9;
    int n = idx & (KDIM - 1);
    int j = n & (QDIM - 1);
    int part = n >> 8;                   // 0 = re, 1 = im
    const float* cs = &csl[m * 16];
    float prod = 1.0f;
    #pragma unroll
    for (int w = 0; w < NQ; ++w) {
      int bit = (j >> (7 - w)) & 1;
      prod *= cs[w * 2 + bit];
    }
    int p = __popc(j) & 3;               // (-i)^p
    float re = (p == 0) ? prod : ((p == 2) ? -prod : 0.0f);
    float im = (p == 1) ? -prod : ((p == 3) ? prod : 0.0f);
    Asub[idx] = part ? im : re;
  }
  __syncthreads();

  // Wave `wave` owns cols [wave*64, wave*64+64), all 64 rows: 4x4 16x16 tiles.
  int ns0 = wave * 64;
  int hl  = lane >> 4;                   // 0: K pair {0,1}, 1: K pair {2,3}
  int lid = lane & 15;
  int koff = hl * 2;

  v8f acc[4][4];
  #pragma unroll
  for (int rs = 0; rs < 4; ++rs)
    #pragma unroll
    for (int ct = 0; ct < 4; ++ct)
      acc[rs][ct] = {};

  for (int kk = 0; kk < KDIM; kk += 4) {
    int kr = kk + koff;
    v2f afrag[4];
    #pragma unroll
    for (int rs = 0; rs < 4; ++rs)
      afrag[rs] = *reinterpret_cast<const v2f*>(&Asub[(rs * 16 + lid) * KDIM + kr]);
    v2f bfrag[4];
    #pragma unroll
    for (int ct = 0; ct < 4; ++ct) {
      int n = ns0 + ct * 16 + lid;
      v2f b;
      b.x = Bmat[kr * NDIM + n];
      b.y = Bmat[(kr + 1) * NDIM + n];
      bfrag[ct] = b;
    }
    #pragma unroll
    for (int rs = 0; rs < 4; ++rs)
      #pragma unroll
      for (int ct = 0; ct < 4; ++ct)
        acc[rs][ct] = __builtin_amdgcn_wmma_f32_16x16x4_f32(
            false, afrag[rs], false, bfrag[ct],
            (short)0, acc[rs][ct], false, false);
  }

  // Epilogue: y[m] += sign(col) * t^2 ; sign flips every 128 realified cols.
  #pragma unroll
  for (int ct = 0; ct < 4; ++ct) {
    int n0 = ns0 + ct * 16;
    float sgn = ((n0 >> 7) & 1) ? -1.0f : 1.0f;
    #pragma unroll
    for (int rs = 0; rs < 4; ++rs) {
      #pragma unroll
      for (int rr = 0; rr < 8; ++rr) {
        float v  = acc[rs][ct][rr];
        float pv = sgn * v * v;
        #pragma unroll
        for (int off = 8; off >= 1; off >>= 1)
          pv += __shfl_xor(pv, off, 32);
        if (lid == 0) {
          int row = rs * 16 + rr + hl * 8;   // C layout: VGPR r -> M=r / M=r+8
          atomicAdd(&yacc[row], pv);
        }
      }
    }
  }
  __syncthreads();
  if (tid < TILE_M) out[m0 + tid] = yacc[tid];
}

// ---------------- launcher ----------------------------------------------------
extern "C" void kernel_launch(void* const* d_in, const int* in_sizes, int n_in,
                              void* d_out, int out_size, void* d_ws, size_t ws_size,
                              hipStream_t stream) {
  (void)in_sizes; (void)n_in; (void)out_size; (void)ws_size;
  const float* x  = (const float*)d_in[0];
  const float* Wp = (const float*)d_in[1];
  const float* bp = (const float*)d_in[2];
  const float* qw = (const float*)d_in[3];
  float* out = (float*)d_out;

  float* csbuf = (float*)d_ws;              // BATCH*16 floats   = 2 MB
  float* Bmat  = csbuf + BATCH * 16;        // 512*512 floats    = 1 MB

  qf_angles_kernel<<<(BATCH * NQ) / 256, 256, 0, stream>>>(x, Wp, bp, csbuf);
  qf_build_B_kernel<<<QDIM, QDIM, 0, stream>>>(qw, Bmat);
  qf_gemm_kernel<<<BATCH / TILE_M, 256, 0, stream>>>(csbuf, Bmat, out);
}